// PairBiasAttention_38534446580063
// MI455X (gfx1250) — compile-verified
//
#include <hip/hip_runtime.h>
#include <hip/hip_bf16.h>

// ---------------------------------------------------------------------------
// PairBiasAttention for MI455X (gfx1250), wave32 + WMMA f32 16x16x4 + TDM.
//   L=1024, C_S=384, H=12, D=32, C_Z=128
// Roofline: z is 536MB -> ~23us floor @ 23.3TB/s. The z@Wz bias GEMM
// (AI ~6 FLOP/B -> ~140 TF32 needed) runs on v_wmma_f32_16x16x4_f32 with
// A-tiles DMA'd into LDS by the Tensor Data Mover (contiguous 8KB per wave).
// ---------------------------------------------------------------------------

typedef __attribute__((ext_vector_type(2))) float    v2f;
typedef __attribute__((ext_vector_type(8))) float    v8f;
typedef __attribute__((ext_vector_type(4))) unsigned v4u;
typedef __attribute__((ext_vector_type(8))) int      v8i;
typedef __attribute__((ext_vector_type(4))) int      v4i;

#define L_SEQ 1024
#define C_S   384
#define N_H   12
#define H_D   32
#define C_Z   128
#define ZPITCH 132   // 128 + 4 DWORD TDM pad -> conflict-free LDS columns

// ---------------------------------------------------------------------------
// Kernel 1/4: C[1024,384] = A[1024,384] @ W[384,384] + b, via WMMA f32.
// One 16x16 tile per wave; 96 x v_wmma_f32_16x16x4_f32 per tile.
// ---------------------------------------------------------------------------
__global__ void gemm384_wmma(const float* __restrict__ A,
                             const float* __restrict__ W0, const float* __restrict__ b0, float* __restrict__ C0,
                             const float* __restrict__ W1, const float* __restrict__ b1, float* __restrict__ C1,
                             const float* __restrict__ W2, const float* __restrict__ b2, float* __restrict__ C2) {
  const float* W; const float* bvec; float* C;
  if (blockIdx.y == 0)      { W = W0; bvec = b0; C = C0; }
  else if (blockIdx.y == 1) { W = W1; bvec = b1; C = C1; }
  else                      { W = W2; bvec = b2; C = C2; }

  const int tid  = threadIdx.x;
  const int lane = tid & 31;
  const int wave = tid >> 5;
  const int tileId = blockIdx.x * 4 + wave;          // 1536 tiles total
  const int m0 = (tileId / (C_S / 16)) * 16;
  const int n0 = (tileId % (C_S / 16)) * 16;

  const int ma = lane & 15;                // A: M row / B: N col for this lane
  const int hk = (lane >> 4) << 1;         // K sub-offset: 0 / 2

  v8f acc = {};
  for (int kb = 0; kb < C_S; kb += 4) {
    v2f a = *(const v2f*)&A[(size_t)(m0 + ma) * C_S + kb + hk];
    v2f bf;
    bf.x = W[(size_t)(kb + hk)     * C_S + n0 + ma];
    bf.y = W[(size_t)(kb + hk + 1) * C_S + n0 + ma];
    acc = __builtin_amdgcn_wmma_f32_16x16x4_f32(false, a, false, bf,
                                                (short)0, acc, false, false);
  }

  const float bn = bvec[n0 + ma];
#pragma unroll
  for (int r = 0; r < 8; ++r) {
    const int m = r + ((lane >> 4) << 3);  // D layout: VGPR r -> M=r / M=r+8
    C[(size_t)(m0 + m) * C_S + n0 + ma] = acc[r] + bn;
  }
}

// ---------------------------------------------------------------------------
// Kernel 2/4 (dominant): bias[h][i][j] = z[i,j,:] @ Wz[:,h] + bz[h].
// Each wave: TDM-load its contiguous 8KB z tile (16 rows x 128 f32) into LDS
// with 4-DWORD padding every 128 DWORDs (pitch 132 -> bank-conflict-free),
// then 32 x v_wmma_f32_16x16x4_f32 over K=128 from LDS. Result transposed
// through LDS so the head-major bias tensor is written as 512B runs.
// ---------------------------------------------------------------------------
__global__ void biasz_wmma(const float* __restrict__ z,
                           const float* __restrict__ Wz,
                           const float* __restrict__ bz,
                           float* __restrict__ biasH) {
  __shared__ float wls[C_Z * 16];            // Wz padded to [128][16]   (8KB)
  __shared__ float tls[128 * 17];            // transpose staging       (8.5KB)
  __shared__ float ztile[8][16 * ZPITCH];    // per-wave TDM dest       (66KB)

  const int tid  = threadIdx.x;              // 256 threads = 8 waves
  const int lane = tid & 31;
  const int wave = tid >> 5;

  const long ij0 = (long)blockIdx.x * 128 + wave * 16;

  // --- Issue TDM load of this wave's 16x128 f32 tile (1-D, 2048 elems) ----
  {
    const unsigned long long ga =
        (unsigned long long)(const void*)z + (unsigned long long)ij0 * (C_Z * 4);
    unsigned g0w0 = 1u;                                        // count=1
    unsigned g0w1 = (unsigned)(unsigned long long)(uintptr_t)&ztile[wave][0];
    unsigned g0w2 = (unsigned)ga;                              // global_addr lo
    unsigned g0w3 = ((unsigned)(ga >> 32) & 0x01FFFFFFu) | (2u << 30); // +type=2
    // group1: mask=0 | data_size=4B | pad_enable | interval=128DW | amount=4DW
    unsigned g1w0 = (2u << 16) | (1u << 20) | (6u << 22) | (3u << 25);
    unsigned g1w1 = (2048u & 0xFFFFu) << 16;   // tensor_dim0 = 2048
    unsigned g1w2 = (1u << 16);                // tensor_dim1 = 1
    unsigned g1w3 = (2048u << 16);             // tile_dim0 = 2048
    unsigned g1w4 = 1u;                        // tile_dim1 = 1
    unsigned g1w5 = 2048u;                     // tensor_dim0_stride
    // Force wave-uniform SGPR values.
    g0w0 = __builtin_amdgcn_readfirstlane(g0w0);
    g0w1 = __builtin_amdgcn_readfirstlane(g0w1);
    g0w2 = __builtin_amdgcn_readfirstlane(g0w2);
    g0w3 = __builtin_amdgcn_readfirstlane(g0w3);
    v4u g0 = { g0w0, g0w1, g0w2, g0w3 };
    v8i g1 = { (int)g1w0, (int)g1w1, (int)g1w2, (int)g1w3,
               (int)g1w4, (int)g1w5, 0, 0 };
    v4i gz = { 0, 0, 0, 0 };
#if defined(__clang_major__) && (__clang_major__ >= 23)
    v8i gz8 = { 0, 0, 0, 0, 0, 0, 0, 0 };
    __builtin_amdgcn_tensor_load_to_lds(g0, g1, gz, gz, gz8, 0);
#else
    __builtin_amdgcn_tensor_load_to_lds(g0, g1, gz, gz, 0);
#endif
  }

  // --- Stage zero-padded Wz while the TDM transfer is in flight ----------
  for (int i = tid; i < C_Z * 16; i += 256) {
    const int kk = i >> 4, n = i & 15;
    wls[i] = (n < N_H) ? Wz[kk * N_H + n] : 0.0f;
  }

  // Speculative prefetch of the next block's z rows (global_prefetch_b8).
  if (blockIdx.x + 1 < gridDim.x)
    __builtin_prefetch(z + (ij0 + 128) * C_Z + lane * 16, 0, 1);

  __syncthreads();
  __builtin_amdgcn_s_wait_tensorcnt(0);      // our tile is resident in LDS

  const int ma = lane & 15;
  const int hk = (lane >> 4) << 1;
  const float* __restrict__ zl = &ztile[wave][0];

  v8f acc = {};
  for (int kb = 0; kb < C_Z; kb += 4) {
    v2f a = *(const v2f*)&zl[ma * ZPITCH + kb + hk];   // ds_load_b64, no conflicts
    v2f bf;
    bf.x = wls[(kb + hk)     * 16 + ma];
    bf.y = wls[(kb + hk + 1) * 16 + ma];
    acc = __builtin_amdgcn_wmma_f32_16x16x4_f32(false, a, false, bf,
                                                (short)0, acc, false, false);
  }

  const float bzn = (ma < N_H) ? bz[ma] : 0.0f;
#pragma unroll
  for (int r = 0; r < 8; ++r) {
    const int m = r + ((lane >> 4) << 3);
    tls[(wave * 16 + m) * 17 + ma] = acc[r] + bzn;
  }
  __syncthreads();

  // Head-major coalesced store: 12 heads x 128 contiguous ij each.
  const long base = (long)blockIdx.x * 128;
  for (int i = tid; i < N_H * 128; i += 256) {
    const int h = i >> 7, row = i & 127;
    biasH[(long)h * ((long)L_SEQ * L_SEQ) + base + row] = tls[row * 17 + h];
  }
}

// ---------------------------------------------------------------------------
// Kernel 3/4: flash attention, one thread per (head, query row).
// ---------------------------------------------------------------------------
__global__ void attn_flash(const float* __restrict__ q,
                           const float* __restrict__ k,
                           const float* __restrict__ v,
                           const float* __restrict__ biasH,
                           const unsigned char* __restrict__ mask,
                           float* __restrict__ o) {
  __shared__ float kt[128 * H_D];
  __shared__ float vt[128 * H_D];

  const int h = blockIdx.y;
  const int l = blockIdx.x * 64 + threadIdx.x;

  const float scale = 0.17677669529663687f;  // 1/sqrt(32)
  float qr[H_D];
#pragma unroll
  for (int d = 0; d < H_D; ++d)
    qr[d] = q[(size_t)l * C_S + h * H_D + d] * scale;

  float mrun = -1e30f, srun = 0.0f;
  float acc[H_D];
#pragma unroll
  for (int d = 0; d < H_D; ++d) acc[d] = 0.0f;

  const float* bh = biasH + (long)h * ((long)L_SEQ * L_SEQ) + (long)l * L_SEQ;
  const unsigned char* mrow = mask + (size_t)l * L_SEQ;

  for (int j0 = 0; j0 < L_SEQ; j0 += 128) {
    __syncthreads();
    for (int idx = threadIdx.x; idx < 128 * (H_D / 4); idx += 64) {
      const int row = idx >> 3;
      const int c4  = (idx & 7) << 2;
      const size_t g = (size_t)(j0 + row) * C_S + h * H_D + c4;
      *(float4*)&kt[row * H_D + c4] = *(const float4*)&k[g];
      *(float4*)&vt[row * H_D + c4] = *(const float4*)&v[g];
    }
    __syncthreads();

    for (int jj = 0; jj < 128; ++jj) {
      const int j = j0 + jj;
      float s = bh[j];
#pragma unroll
      for (int d = 0; d < H_D; ++d) s = fmaf(qr[d], kt[jj * H_D + d], s);
      if (!mrow[j]) s = -1e9f;

      float p;
      if (s > mrun) {                       // lazy rescale (rare path)
        const float c = __expf(mrun - s);
        srun *= c;
#pragma unroll
        for (int d = 0; d < H_D; ++d) acc[d] *= c;
        mrun = s;
        p = 1.0f;
      } else {
        p = __expf(s - mrun);
      }
      srun += p;
#pragma unroll
      for (int d = 0; d < H_D; ++d) acc[d] = fmaf(p, vt[jj * H_D + d], acc[d]);
    }
  }

  const float inv = 1.0f / srun;
#pragma unroll
  for (int d = 0; d < H_D; ++d)
    o[(size_t)l * C_S + h * H_D + d] = acc[d] * inv;
}

// ---------------------------------------------------------------------------
// Launch. Inputs (setup_inputs order):
//  0:s 1:z 2:pair_mask 3:Wq 4:bq 5:Wk 6:bk 7:Wv 8:bv 9:Wz 10:bz 11:Wo 12:bo
// ---------------------------------------------------------------------------
extern "C" void kernel_launch(void* const* d_in, const int* in_sizes, int n_in,
                              void* d_out, int out_size, void* d_ws, size_t ws_size,
                              hipStream_t stream) {
  const float* s    = (const float*)d_in[0];
  const float* z    = (const float*)d_in[1];
  const unsigned char* mask = (const unsigned char*)d_in[2];
  const float* Wq = (const float*)d_in[3];
  const float* bq = (const float*)d_in[4];
  const float* Wk = (const float*)d_in[5];
  const float* bk = (const float*)d_in[6];
  const float* Wv = (const float*)d_in[7];
  const float* bv = (const float*)d_in[8];
  const float* Wz = (const float*)d_in[9];
  const float* bz = (const float*)d_in[10];
  const float* Wo = (const float*)d_in[11];
  const float* bo = (const float*)d_in[12];
  float* out = (float*)d_out;

  // Workspace layout (floats): q | k | v | o | biasH[12][1024][1024]  (~55 MB)
  float* ws   = (float*)d_ws;
  const size_t QKV = (size_t)L_SEQ * C_S;     // 393216
  float* qb    = ws;
  float* kb    = qb + QKV;
  float* vb    = kb + QKV;
  float* ob    = vb + QKV;
  float* biasH = ob + QKV;

  // 1) QKV projections (WMMA): 1536 tiles x 3 matrices, 4 waves/block.
  gemm384_wmma<<<dim3(384, 3), 128, 0, stream>>>(
      s, Wq, bq, qb, Wk, bk, kb, Wv, bv, vb);

  // 2) Pair-bias GEMM over z (dominant, WMMA + TDM): 1M ij rows / 128 per blk.
  biasz_wmma<<<dim3(8192), 256, 0, stream>>>(z, Wz, bz, biasH);

  // 3) Flash attention: 12 heads x 16 row-groups of 64.
  attn_flash<<<dim3(16, N_H), 64, 0, stream>>>(qb, kb, vb, biasH, mask, ob);

  // 4) Output projection (WMMA) into d_out.
  gemm384_wmma<<<dim3(384, 1), 128, 0, stream>>>(
      ob, Wo, bo, out, Wo, bo, out, Wo, bo, out);
}